// ComposedCliffordSteerableKernel_64003602645465
// MI455X (gfx1250) — compile-verified
//
#include <hip/hip_runtime.h>

// ---------------------------------------------------------------------------
// ComposedCliffordSteerableKernel for MI455X (gfx1250, wave32, WMMA)
//
// out[(m*8+bo)*128 + n*8+o, p] =
//     factor * shell[idx] * sum_{j,tau} k1[m*8+bo, n*8+j, p+tau-3] *
//                                       k2[m*8+o , n*8+j, tau]
//
// One workgroup per (m,n) pair (256 workgroups, 512 threads = 16 wave32).
// Both 8x8x343 fp32 tiles (87.8 KB each) are staged into LDS by the Tensor
// Data Mover (3D D# descriptor), then each wave runs a shift-based implicit
// GEMM over the 343 taps using V_WMMA_F32_16X16X4_F32:
//   A (16x4)  rows = 2 output positions x 8 bo,  K = 4 of the 8 j blades
//   B (4x16)  cols = 8 output blades o (cols 8..15 zero),  K = j
//   C (16x16) f32 accumulator, two accumulators to break the WMMA RAW chain
// ---------------------------------------------------------------------------

typedef __attribute__((ext_vector_type(2))) float        v2f;
typedef __attribute__((ext_vector_type(8))) float        v8f;
typedef __attribute__((ext_vector_type(4))) unsigned int u32x4;
typedef __attribute__((ext_vector_type(8))) int          i32x8;
typedef __attribute__((ext_vector_type(4))) int          i32x4;

#define NB    8
#define CI    16
#define CO    16
#define KS    7
#define VOL   343                    // 7*7*7
#define CHS   (CI * NB)              // 128 columns in the full channel grid
#define TILE_ELEMS (NB * NB * VOL)   // 21952 floats per staged (m,n) tile

#if __has_builtin(__builtin_amdgcn_tensor_load_to_lds)
// Issue one TDM load of an 8(bo) x 8(j) x 343 fp32 tile into LDS.
// gbase points at element (m*8, n*8, 0) of a (128,128,343) fp32 tensor.
//   dim0 = 343 contiguous floats (flattened z,y,x)
//   dim1 = j, stride 343 floats
//   dim2 = bo, stride 128*343 = 43904 floats
__device__ __forceinline__ void tdm_load_tile(const float* gbase,
                                              unsigned lds_byte_off) {
  unsigned long long ga = (unsigned long long)(__SIZE_TYPE__)gbase;

  u32x4 g0;
  g0[0] = 1u;                                   // count=1, user descriptor
  g0[1] = lds_byte_off;                         // lds_addr (bytes)
  g0[2] = (unsigned)(ga & 0xFFFFFFFFull);       // global_addr[31:0]
  g0[3] = (unsigned)((ga >> 32) & 0x1FFFFFFull) // global_addr[56:32]
          | 0x80000000u;                        // type = 2 ("image")

  i32x8 g1;
  g1[0] = (2 << 16);                            // data_size = 4 bytes
  g1[1] = (int)((unsigned)VOL << 16);           // tensor_dim0[15:0]
  g1[2] = (int)((unsigned)NB << 16);            // tensor_dim1[15:0]
  g1[3] = (int)((unsigned)VOL << 16);           // tile_dim0 = 343
  g1[4] = NB | (NB << 16);                      // tile_dim1 | tile_dim2
  g1[5] = VOL;                                  // tensor_dim0_stride[31:0]
  g1[6] = (int)(((unsigned)(CHS * VOL) & 0xFFFFu) << 16); // dim1_stride[15:0]
  g1[7] = (int)((unsigned)(CHS * VOL) >> 16);   // dim1_stride[47:16]

  i32x4 g2;
  g2[0] = NB;                                   // tensor_dim2 = 8
  g2[1] = 0; g2[2] = 0; g2[3] = 0;

  i32x4 g3; g3[0] = 0; g3[1] = 0; g3[2] = 0; g3[3] = 0;

#if __clang_major__ >= 23
  i32x8 g4;
  g4[0]=0; g4[1]=0; g4[2]=0; g4[3]=0; g4[4]=0; g4[5]=0; g4[6]=0; g4[7]=0;
  __builtin_amdgcn_tensor_load_to_lds(g0, g1, g2, g3, g4, 0);
#else
  __builtin_amdgcn_tensor_load_to_lds(g0, g1, g2, g3, 0);
#endif
}
#endif  // has tensor_load_to_lds

__global__ __launch_bounds__(512, 1)
void clifford_compose_wmma(const float* __restrict__ k1,
                           const float* __restrict__ k2,
                           const float* __restrict__ shell,
                           const float* __restrict__ factor,
                           float* __restrict__ out) {
  extern __shared__ float smem[];
  float* k1s = smem;               // [bo][j][343]
  float* k2s = smem + TILE_ELEMS;  // [o ][j][343]

  const int mn = blockIdx.x;
  const int m  = mn >> 4;
  const int n  = mn & 15;
  const int chanBase = ((m * NB) * CHS + n * NB) * VOL;  // element offset

  const int tid  = threadIdx.x;
  const int wave = tid >> 5;
  const int lane = tid & 31;

  // ---- Stage both tiles into LDS ----------------------------------------
#if __has_builtin(__builtin_amdgcn_tensor_load_to_lds)
  if (wave == 0) {
    tdm_load_tile(k1 + chanBase, 0u);
    tdm_load_tile(k2 + chanBase, (unsigned)(TILE_ELEMS * sizeof(float)));
    __builtin_amdgcn_s_wait_tensorcnt(0);
  }
#else
  for (int i = tid; i < TILE_ELEMS; i += 512) {
    const int ch = i / VOL;           // bo*8 + j
    const int sp = i - ch * VOL;
    const int boI = ch >> 3, jI = ch & 7;
    const int g = chanBase + (boI * CHS + jI) * VOL + sp;
    k1s[i] = k1[g];
    k2s[i] = k2[g];
  }
#endif
  __syncthreads();

  const float fac = factor[0];

  // ---- Per-lane WMMA fragment geometry (wave32) -------------------------
  // A 16x4 fp32: lanes 0-15 hold M=0..15 with K={0,1}; lanes 16-31 K={2,3}.
  const int Mrow = lane & 15;            // A row
  const int bo   = Mrow & 7;             // output blade of k1
  const int sA   = Mrow >> 3;            // which of the 2 positions
  const int kb   = (lane >> 4) << 1;     // K base (0 or 2) within 4-wide step
  // B 4x16: lane N = lane&15, same K split; columns 8..15 are zero padding.
  const int ocl    = lane & 7;           // clamped B column (o)
  const bool oVal  = (lane & 15) < 8;

  // 172 position-pair tiles cover p = 0..342 (last pair is half valid).
  for (int tile = wave; tile < 172; tile += 16) {
    const int  p0  = tile * 2;
    const int  pA  = p0 + sA;
    const bool pAv = pA < VOL;
    const int  pAc = pAv ? pA : 0;
    const int  zA  = pAc / 49;
    const int  rA  = pAc - zA * 49;
    const int  yA  = rA / 7;
    const int  xA  = rA - yA * 7;

    v8f c0 = {0.f, 0.f, 0.f, 0.f, 0.f, 0.f, 0.f, 0.f};  // j-steps 0..3
    v8f c1 = {0.f, 0.f, 0.f, 0.f, 0.f, 0.f, 0.f, 0.f};  // j-steps 4..7

    for (int dz = 0; dz < KS; ++dz) {
      const int  sz = zA + dz - 3;
      const bool vz = pAv & (sz >= 0) & (sz < KS);
      for (int dy = 0; dy < KS; ++dy) {
        const int  sy  = yA + dy - 3;
        const bool vzy = vz & (sy >= 0) & (sy < KS);
        const int  sb  = (sz * 7 + sy) * 7;
        const int  tb  = (dz * 7 + dy) * 7;
        for (int dx = 0; dx < KS; ++dx) {
          const int  sx  = xA + dx - 3;
          const bool val = vzy & (sx >= 0) & (sx < KS);
          int sidx = sb + sx;
          sidx = sidx < 0 ? 0 : (sidx > (VOL - 1) ? (VOL - 1) : sidx);
          const int t = tb + dx;

          // K-step 0: j = kb, kb+1
          {
            const int j0 = kb;
            float a0 = k1s[(bo * NB + j0) * VOL + sidx];
            float a1 = k1s[(bo * NB + j0 + 1) * VOL + sidx];
            float b0 = k2s[(ocl * NB + j0) * VOL + t];
            float b1 = k2s[(ocl * NB + j0 + 1) * VOL + t];
            v2f a, b;
            a.x = val ? a0 : 0.f;  a.y = val ? a1 : 0.f;
            b.x = oVal ? b0 : 0.f; b.y = oVal ? b1 : 0.f;
            c0 = __builtin_amdgcn_wmma_f32_16x16x4_f32(
                     false, a, false, b, (short)0, c0, false, false);
          }
          // K-step 1: j = 4+kb, 4+kb+1  (independent accumulator)
          {
            const int j0 = 4 + kb;
            float a0 = k1s[(bo * NB + j0) * VOL + sidx];
            float a1 = k1s[(bo * NB + j0 + 1) * VOL + sidx];
            float b0 = k2s[(ocl * NB + j0) * VOL + t];
            float b1 = k2s[(ocl * NB + j0 + 1) * VOL + t];
            v2f a, b;
            a.x = val ? a0 : 0.f;  a.y = val ? a1 : 0.f;
            b.x = oVal ? b0 : 0.f; b.y = oVal ? b1 : 0.f;
            c1 = __builtin_amdgcn_wmma_f32_16x16x4_f32(
                     false, a, false, b, (short)0, c1, false, false);
          }
        }
      }
    }

    // ---- Epilogue: C 16x16 layout -> lanes 0-15 rows 0-7, 16-31 rows 8-15.
    // Row M = v + 8*(lane>>4): position selector sC = lane>>4, bo = v.
    const int oCol = lane & 15;
    const int sC   = lane >> 4;
    const int p    = p0 + sC;
    if (oCol < 8 && p < VOL) {
      const int colBase = n * NB + oCol;
#pragma unroll
      for (int v = 0; v < 8; ++v) {
        const int idx = ((m * NB + v) * CHS + colBase) * VOL + p;
        out[idx] = (c0[v] + c1[v]) * shell[idx] * fac;
      }
    }
  }
}

extern "C" void kernel_launch(void* const* d_in, const int* in_sizes, int n_in,
                              void* d_out, int out_size, void* d_ws,
                              size_t ws_size, hipStream_t stream) {
  (void)in_sizes; (void)n_in; (void)d_ws; (void)ws_size; (void)out_size;
  const float* k1     = (const float*)d_in[0];
  const float* k2     = (const float*)d_in[1];
  const float* shell  = (const float*)d_in[2];
  const float* factor = (const float*)d_in[3];
  float* out = (float*)d_out;

  dim3 grid(CO * CI);   // 256 (m,n) pairs
  dim3 block(512);      // 16 wave32
  size_t shmem = 2u * TILE_ELEMS * sizeof(float);  // 175,616 B of 320 KB LDS
  hipLaunchKernelGGL(clifford_compose_wmma, grid, block, shmem, stream,
                     k1, k2, shell, factor, out);
}